// HAN2_52467320487956
// MI455X (gfx1250) — compile-verified
//
#include <hip/hip_runtime.h>
#include <hip/hip_bf16.h>

typedef float v2f __attribute__((ext_vector_type(2)));
typedef float v8f __attribute__((ext_vector_type(8)));

#define NAUT 50000
#define NPAP 100000
#define NEDG 300000
#define HIDC 256
#define OUTC 64

__device__ __forceinline__ unsigned f2ord(float f) {
  unsigned u = __float_as_uint(f);
  return (u & 0x80000000u) ? ~u : (u | 0x80000000u);
}
__device__ __forceinline__ float ord2f(unsigned u) {
  return (u & 0x80000000u) ? __uint_as_float(u & 0x7fffffffu) : __uint_as_float(~u);
}

// ---------------- elementwise helpers ----------------
__global__ void fill_f32(float* __restrict__ p, float v, long n) {
  long i = (long)blockIdx.x * blockDim.x + threadIdx.x;
  if (i < n) p[i] = v;
}
__global__ void fill_u32(unsigned* __restrict__ p, unsigned v, long n) {
  long i = (long)blockIdx.x * blockDim.x + threadIdx.x;
  if (i < n) p[i] = v;
}
__global__ void relu_ip(float* __restrict__ p, long n) {
  long i = (long)blockIdx.x * blockDim.x + threadIdx.x;
  if (i < n) p[i] = fmaxf(p[i], 0.0f);
}
__global__ void copy_f32(const float* __restrict__ s, float* __restrict__ d, long n) {
  long i = (long)blockIdx.x * blockDim.x + threadIdx.x;
  if (i < n) d[i] = s[i];
}

// ---------------- WMMA fp32 GEMM: C[M,N] = act(A[M,K] @ W[K,N] + b) ----------------
// One wave32 computes a 64x16 tile (4 stacked 16x16 M-subtiles) via
// V_WMMA_F32_16X16X4_F32, reusing one B fragment across 4 WMMAs per K-step.
// B (weights) is L2-resident; A is the streamed operand at ~8B/lane/WMMA.
// A layout (16x4 f32): lane L holds row M=L&15; lane-half picks K pair (0,1)/(2,3).
// B layout (4x16 f32): lane L holds col N=L&15; lane-half picks K rows (0,1)/(2,3).
// C/D (16x16 f32): lane L holds col N=L&15; VGPR v holds row v + 8*(L>=16).
// ACT: 0 = none, 1 = ELU.  COLSUM: fuse tanh + per-column sum (no C store).
template <int ACT, bool COLSUM>
__global__ void gemm64x16(const float* __restrict__ A, const float* __restrict__ W,
                          const float* __restrict__ bias, float* __restrict__ C,
                          float* __restrict__ colsum, int M, int K, int N) {
  const int wavesPerBlk = blockDim.x >> 5;
  int wave = blockIdx.x * wavesPerBlk + ((int)threadIdx.x >> 5);
  int tilesN = N >> 4;
  int mtiles = M >> 4;                 // M is always a multiple of 16 here
  int mgroups = (mtiles + 3) >> 2;
  if (wave >= mgroups * tilesN) return;  // wave-uniform: EXEC stays all-ones for WMMA
  int tg = wave / tilesN;
  int tn = wave - tg * tilesN;
  int lane = threadIdx.x & 31;
  int l15 = lane & 15;
  int half = lane >> 4;
  int koff = half << 1;
  int n = (tn << 4) + l15;
  int mt0 = tg << 2;
  // clamp subtile rows for the (rare) partial last M-group; stores are guarded
  int mt1 = mt0 + 1 < mtiles ? mt0 + 1 : mtiles - 1;
  int mt2 = mt0 + 2 < mtiles ? mt0 + 2 : mtiles - 1;
  int mt3 = mt0 + 3 < mtiles ? mt0 + 3 : mtiles - 1;
  const float* ar0 = A + ((long)(mt0 << 4) + l15) * K + koff;
  const float* ar1 = A + ((long)(mt1 << 4) + l15) * K + koff;
  const float* ar2 = A + ((long)(mt2 << 4) + l15) * K + koff;
  const float* ar3 = A + ((long)(mt3 << 4) + l15) * K + koff;
  const float* wcol = W + (long)koff * N + n;
  v8f acc0 = {0.f, 0.f, 0.f, 0.f, 0.f, 0.f, 0.f, 0.f};
  v8f acc1 = acc0, acc2 = acc0, acc3 = acc0;
#pragma unroll 2
  for (int k = 0; k < K; k += 4) {
    v2f b;
    b.x = wcol[(long)k * N];
    b.y = wcol[(long)(k + 1) * N];
    v2f a0 = *(const v2f*)(ar0 + k);   // 8B aligned: k+koff even, rows even-length
    v2f a1 = *(const v2f*)(ar1 + k);
    v2f a2 = *(const v2f*)(ar2 + k);
    v2f a3 = *(const v2f*)(ar3 + k);
    acc0 = __builtin_amdgcn_wmma_f32_16x16x4_f32(false, a0, false, b, (short)0, acc0, false, false);
    acc1 = __builtin_amdgcn_wmma_f32_16x16x4_f32(false, a1, false, b, (short)0, acc1, false, false);
    acc2 = __builtin_amdgcn_wmma_f32_16x16x4_f32(false, a2, false, b, (short)0, acc2, false, false);
    acc3 = __builtin_amdgcn_wmma_f32_16x16x4_f32(false, a3, false, b, (short)0, acc3, false, false);
  }
  float bn = bias[n];
  if (!COLSUM) {
    int rh = half << 3;
#pragma unroll
    for (int s = 0; s < 4; ++s) {
      if (mt0 + s >= mtiles) break;
      v8f acc = s == 0 ? acc0 : (s == 1 ? acc1 : (s == 2 ? acc2 : acc3));
      int row0 = ((mt0 + s) << 4) + rh;
#pragma unroll
      for (int v = 0; v < 8; ++v) {
        float val = acc[v] + bn;
        if (ACT == 1) val = val > 0.0f ? val : (__expf(val) - 1.0f);
        C[(long)(row0 + v) * N + n] = val;
      }
    }
  } else {
    float ssum = 0.0f;
#pragma unroll
    for (int s = 0; s < 4; ++s) {
      if (mt0 + s >= mtiles) break;
      v8f acc = s == 0 ? acc0 : (s == 1 ? acc1 : (s == 2 ? acc2 : acc3));
#pragma unroll
      for (int v = 0; v < 8; ++v) ssum += tanhf(acc[v] + bn);
    }
    atomicAdd(&colsum[n], ssum);       // lane-halves/subtiles fold rows via atomics
  }
}

// ---------------- HAN edge-type pipeline ----------------
// a[n,h] = dot(x[n,h,:], att[h,:])
__global__ void attn_dot(const float* __restrict__ x, const float* __restrict__ att,
                         float* __restrict__ out, int Nn, int H, int D) {
  int i = blockIdx.x * blockDim.x + threadIdx.x;
  if (i >= Nn * H) return;
  int n = i / H, h = i - n * H;
  const float* xr = x + ((long)n * H + h) * D;
  const float* ar = att + (long)h * D;
  float s = 0.f;
  for (int d = 0; d < D; ++d) s += xr[d] * ar[d];
  out[i] = s;
}

// alpha = leaky_relu(a_s[src]+a_d[dst], 0.2); segment-max via ordered-uint atomicMax
__global__ void edge_alpha(const int* __restrict__ esrc, const int* __restrict__ edst,
                           const float* __restrict__ as_, const float* __restrict__ ad_,
                           float* __restrict__ alpha, unsigned* __restrict__ amax,
                           int E_, int H) {
  int i = blockIdx.x * blockDim.x + threadIdx.x;
  if (i >= E_ * H) return;
  int e = i / H, h = i - e * H;
  int di = edst[e] * H + h;
  float v = as_[esrc[e] * H + h] + ad_[di];
  v = v > 0.f ? v : 0.2f * v;
  alpha[i] = v;
  atomicMax(&amax[di], f2ord(v));
}

// ex = exp(alpha - amax[dst]) (in place); denom[dst] += ex
__global__ void edge_exp(const int* __restrict__ edst, float* __restrict__ alpha,
                         const unsigned* __restrict__ amax, float* __restrict__ denom,
                         int E_, int H) {
  int i = blockIdx.x * blockDim.x + threadIdx.x;
  if (i >= E_ * H) return;
  int e = i / H, h = i - e * H;
  int di = edst[e] * H + h;
  float v = __expf(alpha[i] - ord2f(amax[di]));
  alpha[i] = v;
  atomicAdd(&denom[di], v);
}

// out[dst,h,:] += xs[src,h,:] * ex/(denom[dst,h]+1e-16); one wave per (edge,head)
__global__ void edge_scatter(const int* __restrict__ esrc, const int* __restrict__ edst,
                             const float* __restrict__ ex, const float* __restrict__ denom,
                             const float* __restrict__ xs, float* __restrict__ out,
                             int E_, int H, int D) {
  int g = blockIdx.x * blockDim.x + threadIdx.x;
  int widx = g >> 5, lane = g & 31;
  if (widx >= E_ * H) return;
  int e = widx / H, h = widx - e * H;
  int s = esrc[e], d0 = edst[e];
  float w = ex[widx] / (denom[d0 * H + h] + 1e-16f);
  const float* xrow = xs + ((long)s * H + h) * D;
  float* orow = out + ((long)d0 * H + h) * D;
  for (int d = lane; d < D; d += 32) atomicAdd(&orow[d], xrow[d] * w);
}

// ---------------- semantic attention (M=2 case) ----------------
__global__ void sem_score(const float* __restrict__ cs0, const float* __restrict__ cs1,
                          const float* __restrict__ q, float invN, int C,
                          float* __restrict__ score) {
  __shared__ float red[256];
  const float* cs = blockIdx.x ? cs1 : cs0;
  float s = 0.f;
  for (int c = threadIdx.x; c < C; c += blockDim.x) s += cs[c] * q[c];
  red[threadIdx.x] = s;
  __syncthreads();
  for (int o = blockDim.x >> 1; o > 0; o >>= 1) {
    if ((int)threadIdx.x < o) red[threadIdx.x] += red[threadIdx.x + o];
    __syncthreads();
  }
  if (threadIdx.x == 0) score[blockIdx.x] = red[0] * invN;
}
__global__ void softmax2(float* s) {
  float a = s[0], b = s[1], mx = fmaxf(a, b);
  float ea = __expf(a - mx), eb = __expf(b - mx);
  float inv = 1.0f / (ea + eb);
  s[0] = ea * inv;
  s[1] = eb * inv;
}
__global__ void combine2(const float* s0, const float* __restrict__ s1,
                         const float* __restrict__ attn, float* dst, long n) {
  long i = (long)blockIdx.x * blockDim.x + threadIdx.x;
  if (i < n) dst[i] = attn[0] * s0[i] + attn[1] * s1[i];
}

// ---------------- host-side helpers ----------------
static inline int cdiv(long a, long b) { return (int)((a + b - 1) / b); }

static void run_gemm(int act, bool colsum, const float* A, const float* W, const float* b,
                     float* C, float* cs, int M, int K, int N, hipStream_t st) {
  int mgroups = (M / 16 + 3) / 4;
  int waves = mgroups * (N / 16);
  int blocks = (waves + 7) / 8;
  if (colsum)
    gemm64x16<0, true><<<blocks, 256, 0, st>>>(A, W, b, C, cs, M, K, N);
  else if (act == 1)
    gemm64x16<1, false><<<blocks, 256, 0, st>>>(A, W, b, C, cs, M, K, N);
  else
    gemm64x16<0, false><<<blocks, 256, 0, st>>>(A, W, b, C, cs, M, K, N);
}

static void run_edge(const int* ei, int ne, const float* xs, int Ns, const float* xd, int Nd,
                     const float* att_s, const float* att_d, float* out, float* sAS, float* sAD,
                     unsigned* sAMAX, float* sDEN, float* sALPHA, int H, int D, hipStream_t st) {
  const int* esrc = ei;
  const int* edst = ei + ne;
  const int T = 256;
  attn_dot<<<cdiv((long)Ns * H, T), T, 0, st>>>(xs, att_s, sAS, Ns, H, D);
  attn_dot<<<cdiv((long)Nd * H, T), T, 0, st>>>(xd, att_d, sAD, Nd, H, D);
  fill_u32<<<cdiv((long)Nd * H, T), T, 0, st>>>(sAMAX, 0x007FFFFFu /*ord(-inf)*/, (long)Nd * H);
  fill_f32<<<cdiv((long)Nd * H, T), T, 0, st>>>(sDEN, 0.f, (long)Nd * H);
  edge_alpha<<<cdiv((long)ne * H, T), T, 0, st>>>(esrc, edst, sAS, sAD, sALPHA, sAMAX, ne, H);
  edge_exp<<<cdiv((long)ne * H, T), T, 0, st>>>(edst, sALPHA, sAMAX, sDEN, ne, H);
  edge_scatter<<<cdiv((long)ne * H * 32, T), T, 0, st>>>(esrc, edst, sALPHA, sDEN, xs, out, ne, H, D);
}

extern "C" void kernel_launch(void* const* d_in, const int* in_sizes, int n_in,
                              void* d_out, int out_size, void* d_ws, size_t ws_size,
                              hipStream_t stream) {
  if (n_in < 35) return;
  const float* x_author = (const float*)d_in[0];
  const float* x_paper  = (const float*)d_in[1];
  const int* ei_writes  = (const int*)d_in[2];
  const int* ei_wb      = (const int*)d_in[3];
  const int* ei_cites   = (const int*)d_in[4];
  // params pytree, sorted-key flatten: conv1, conv2, lin
  const float* c1_ad_wp = (const float*)d_in[5];   // att_dst author__writes__paper
  const float* c1_ad_cp = (const float*)d_in[6];   // att_dst paper__cites__paper
  const float* c1_ad_wb = (const float*)d_in[7];   // att_dst paper__written_by__author
  const float* c1_as_wp = (const float*)d_in[8];
  const float* c1_as_cp = (const float*)d_in[9];
  const float* c1_as_wb = (const float*)d_in[10];
  const float* c1_kb    = (const float*)d_in[11];
  const float* c1_kw    = (const float*)d_in[12];
  const float* c1_pa_b  = (const float*)d_in[13];
  const float* c1_pa_w  = (const float*)d_in[14];
  const float* c1_pp_b  = (const float*)d_in[15];
  const float* c1_pp_w  = (const float*)d_in[16];
  const float* c1_q     = (const float*)d_in[17];
  const float* c2_ad_wp = (const float*)d_in[18];
  const float* c2_ad_cp = (const float*)d_in[19];
  const float* c2_ad_wb = (const float*)d_in[20];
  const float* c2_as_wp = (const float*)d_in[21];
  const float* c2_as_cp = (const float*)d_in[22];
  const float* c2_as_wb = (const float*)d_in[23];
  const float* c2_kb    = (const float*)d_in[24];
  const float* c2_kw    = (const float*)d_in[25];
  const float* c2_pa_b  = (const float*)d_in[26];
  const float* c2_pa_w  = (const float*)d_in[27];
  const float* c2_pp_b  = (const float*)d_in[28];
  const float* c2_pp_w  = (const float*)d_in[29];
  const float* c2_q     = (const float*)d_in[30];
  const float* lin_a_b  = (const float*)d_in[31];
  const float* lin_a_w  = (const float*)d_in[32];
  const float* lin_p_b  = (const float*)d_in[33];
  const float* lin_p_w  = (const float*)d_in[34];

  // workspace bump allocation with lifetime reuse
  float* ws = (float*)d_ws;
  long off = 0;
  float* bufA = ws + off; off += (long)NAUT * HIDC;  // h_author -> out_wa1 / res_author1
  float* bufB = ws + off; off += (long)NPAP * HIDC;  // h_paper  -> out_wp1 / res_paper1
  float* bufC = ws + off; off += (long)NAUT * HIDC;  // xa1 -> {xa2, out_wa2}
  float* bufD = ws + off; off += (long)NPAP * HIDC;  // xp1 -> {xp2, out_wp2, out_cp2}
  float* bufE = ws + off; off += (long)NPAP * HIDC;  // out_cp1
  float* sAS  = ws + off; off += (long)NPAP * 2;
  float* sAD  = ws + off; off += (long)NPAP * 2;
  unsigned* sAMAX = (unsigned*)(ws + off); off += (long)NPAP * 2;
  float* sDEN = ws + off; off += (long)NPAP * 2;
  float* sALP = ws + off; off += (long)NEDG * 2;
  float* sCS0 = ws + off; off += HIDC;
  float* sCS1 = ws + off; off += HIDC;
  float* sATT = ws + off; off += 2;

  const int T = 256;

  // ---- layer 0: lin + ELU ----
  run_gemm(1, false, x_author, lin_a_w, lin_a_b, bufA, nullptr, NAUT, 128, HIDC, stream);
  run_gemm(1, false, x_paper,  lin_p_w, lin_p_b, bufB, nullptr, NPAP, HIDC, HIDC, stream);

  // ---- conv1 (heads=2, d=128) ----
  run_gemm(0, false, bufA, c1_pa_w, c1_pa_b, bufC, nullptr, NAUT, HIDC, HIDC, stream);
  run_gemm(0, false, bufB, c1_pp_w, c1_pp_b, bufD, nullptr, NPAP, HIDC, HIDC, stream);
  // h_* dead -> reuse as message accumulators
  fill_f32<<<cdiv((long)NAUT * HIDC, T), T, 0, stream>>>(bufA, 0.f, (long)NAUT * HIDC);
  fill_f32<<<cdiv((long)NPAP * HIDC, T), T, 0, stream>>>(bufB, 0.f, (long)NPAP * HIDC);
  fill_f32<<<cdiv((long)NPAP * HIDC, T), T, 0, stream>>>(bufE, 0.f, (long)NPAP * HIDC);
  run_edge(ei_writes, NEDG, bufC, NAUT, bufD, NPAP, c1_as_wp, c1_ad_wp, bufB,
           sAS, sAD, sAMAX, sDEN, sALP, 2, 128, stream);
  run_edge(ei_wb,     NEDG, bufD, NPAP, bufC, NAUT, c1_as_wb, c1_ad_wb, bufA,
           sAS, sAD, sAMAX, sDEN, sALP, 2, 128, stream);
  run_edge(ei_cites,  NEDG, bufD, NPAP, bufD, NPAP, c1_as_cp, c1_ad_cp, bufE,
           sAS, sAD, sAMAX, sDEN, sALP, 2, 128, stream);
  relu_ip<<<cdiv((long)NAUT * HIDC, T), T, 0, stream>>>(bufA, (long)NAUT * HIDC);
  relu_ip<<<cdiv((long)NPAP * HIDC, T), T, 0, stream>>>(bufB, (long)NPAP * HIDC);
  relu_ip<<<cdiv((long)NPAP * HIDC, T), T, 0, stream>>>(bufE, (long)NPAP * HIDC);
  // semantic attention over paper outputs (writes vs cites); author has M=1 -> identity
  fill_f32<<<1, HIDC, 0, stream>>>(sCS0, 0.f, HIDC);
  fill_f32<<<1, HIDC, 0, stream>>>(sCS1, 0.f, HIDC);
  run_gemm(0, true, bufB, c1_kw, c1_kb, nullptr, sCS0, NPAP, HIDC, HIDC, stream);
  run_gemm(0, true, bufE, c1_kw, c1_kb, nullptr, sCS1, NPAP, HIDC, HIDC, stream);
  sem_score<<<2, 256, 0, stream>>>(sCS0, sCS1, c1_q, 1.0f / (float)NPAP, HIDC, sATT);
  softmax2<<<1, 1, 0, stream>>>(sATT);
  combine2<<<cdiv((long)NPAP * HIDC, T), T, 0, stream>>>(bufB, bufE, sATT, bufB, (long)NPAP * HIDC);
  // res_author1 = bufA, res_paper1 = bufB

  // ---- conv2 (heads=1, d=64) ----
  float* xa2 = bufC;
  float* oWA2 = bufC + (long)NAUT * OUTC;
  float* xp2 = bufD;
  float* oWP2 = bufD + (long)NPAP * OUTC;
  float* oCP2 = bufD + (long)NPAP * OUTC * 2;
  run_gemm(0, false, bufA, c2_pa_w, c2_pa_b, xa2, nullptr, NAUT, HIDC, OUTC, stream);
  run_gemm(0, false, bufB, c2_pp_w, c2_pp_b, xp2, nullptr, NPAP, HIDC, OUTC, stream);
  fill_f32<<<cdiv((long)NAUT * OUTC, T), T, 0, stream>>>(oWA2, 0.f, (long)NAUT * OUTC);
  fill_f32<<<cdiv((long)NPAP * OUTC, T), T, 0, stream>>>(oWP2, 0.f, (long)NPAP * OUTC);
  fill_f32<<<cdiv((long)NPAP * OUTC, T), T, 0, stream>>>(oCP2, 0.f, (long)NPAP * OUTC);
  run_edge(ei_writes, NEDG, xa2, NAUT, xp2, NPAP, c2_as_wp, c2_ad_wp, oWP2,
           sAS, sAD, sAMAX, sDEN, sALP, 1, 64, stream);
  run_edge(ei_wb,     NEDG, xp2, NPAP, xa2, NAUT, c2_as_wb, c2_ad_wb, oWA2,
           sAS, sAD, sAMAX, sDEN, sALP, 1, 64, stream);
  run_edge(ei_cites,  NEDG, xp2, NPAP, xp2, NPAP, c2_as_cp, c2_ad_cp, oCP2,
           sAS, sAD, sAMAX, sDEN, sALP, 1, 64, stream);
  relu_ip<<<cdiv((long)NAUT * OUTC, T), T, 0, stream>>>(oWA2, (long)NAUT * OUTC);
  relu_ip<<<cdiv((long)NPAP * OUTC, T), T, 0, stream>>>(oWP2, (long)NPAP * OUTC);
  relu_ip<<<cdiv((long)NPAP * OUTC, T), T, 0, stream>>>(oCP2, (long)NPAP * OUTC);
  fill_f32<<<1, OUTC, 0, stream>>>(sCS0, 0.f, OUTC);
  fill_f32<<<1, OUTC, 0, stream>>>(sCS1, 0.f, OUTC);
  run_gemm(0, true, oWP2, c2_kw, c2_kb, nullptr, sCS0, NPAP, OUTC, OUTC, stream);
  run_gemm(0, true, oCP2, c2_kw, c2_kb, nullptr, sCS1, NPAP, OUTC, OUTC, stream);
  sem_score<<<2, 256, 0, stream>>>(sCS0, sCS1, c2_q, 1.0f / (float)NPAP, OUTC, sATT);
  softmax2<<<1, 1, 0, stream>>>(sATT);

  float* out_author = (float*)d_out;
  float* out_paper  = (float*)d_out + (long)NAUT * OUTC;
  combine2<<<cdiv((long)NPAP * OUTC, T), T, 0, stream>>>(oWP2, oCP2, sATT, out_paper,
                                                         (long)NPAP * OUTC);
  copy_f32<<<cdiv((long)NAUT * OUTC, T), T, 0, stream>>>(oWA2, out_author, (long)NAUT * OUTC);
}